// EMA_69389491634381
// MI455X (gfx1250) — compile-verified
//
#include <hip/hip_runtime.h>

typedef __attribute__((ext_vector_type(2))) float v2f;
typedef __attribute__((ext_vector_type(4))) float v4f;
typedef __attribute__((ext_vector_type(8))) float v8f;

// Problem dims (match reference)
constexpr int L = 8192;
constexpr int D = 512;
constexpr int E = 8;

constexpr int TILE       = 16;   // rows per WMMA tile
constexpr int NTILES     = 64;   // iterations per block (equal for all blocks)
constexpr int WARM_TILES = 32;   // 512 warm-up rows: a_max^512 ~ 4e-10 < f32 eps
constexpr int SEG0_ROWS  = 1024; // segment 0: no warm-up, stores 64 tiles
constexpr int SEG_ROWS   = 512;  // other segments: 32 warm + 32 stored tiles
constexpr int NSEG       = 1 + (L - SEG0_ROWS) / SEG_ROWS; // 15

// Raw workgroup split-barrier: only LDS visibility is required between waves
// (all cross-thread handoff is through LDS). Avoids __syncthreads()'s implicit
// s_wait_storecnt 0, so NT global stores stream without stalling the loop.
#define WG_BARRIER()                     \
    asm volatile("s_wait_dscnt 0x0\n\t"  \
                 "s_barrier_signal -1\n\t" \
                 "s_barrier_wait -1" ::: "memory")

// grid = (D/16, NSEG) = (32, 15); block = 256 (8 waves, one per e)
__global__ __launch_bounds__(256) void ema_scan_wmma(
    const float* __restrict__ x, const float* __restrict__ log_decay,
    float* __restrict__ out)
{
    __shared__ float ldsX[TILE * 16];            // 16x16 f32 x tile (1 KB)
    __shared__ float ldsOut[2][TILE * 16 * E];   // double-buffered 16x128 (16 KB)

    const int tid  = threadIdx.x;
    const int lane = tid & 31;
    const int e    = tid >> 5;     // wave id == EMA index
    const int half = lane >> 4;    // 0: lanes 0-15, 1: lanes 16-31
    const int n    = lane & 15;    // column (d) within tile / row (m) for A

    const int d0  = blockIdx.x * 16;
    const int seg = blockIdx.y;

    // decay = sigmoid(log_decay[e]); b = decay; a = 1 - decay
    const float ld    = log_decay[e];
    const float bcoef = 1.0f / (1.0f + __expf(-ld));
    const float a     = 1.0f - bcoef;

    // powers a^0 .. a^16
    float ap[17];
    ap[0] = 1.0f;
    #pragma unroll
    for (int i = 1; i <= 16; ++i) ap[i] = ap[i - 1] * a;

    // A fragments: M[m][k] = (k<=m) ? b*a^(m-k) : 0  (lower triangular)
    // f32 16x4 operand layout: vgpr j, lane-half h -> K = 4*kk + 2*h + j ; M = lane&15
    const int m  = n;
    const int kb = 2 * half;
    v2f aM[4];
    #pragma unroll
    for (int kk = 0; kk < 4; ++kk) {
        const int k0 = 4 * kk + kb;
        aM[kk].x = (k0     <= m) ? bcoef * ap[m - k0]     : 0.0f;
        aM[kk].y = (k0 + 1 <= m) ? bcoef * ap[m - k0 - 1] : 0.0f;
    }

    // carry-injection coefficients: acc row m gets a^(m+1) * carry
    float apc[8];
    #pragma unroll
    for (int i = 0; i < 8; ++i) apc[i] = ap[i + 8 * half + 1];

    int warm, lstart;
    float cval;
    if (seg == 0) {
        warm = 0; lstart = 0;
        cval = x[d0 + n];                         // s_{-1} = x[0,:] (per column)
    } else {
        warm   = WARM_TILES;
        lstart = SEG0_ROWS + (seg - 1) * SEG_ROWS - WARM_TILES * TILE;
        cval   = 0.0f;                            // error < a_max^512 ~ 4e-10
    }

    const int xr = tid >> 4;       // row within tile this thread stages
    const int xc = tid & 15;       // col within tile this thread stages

    // flush one finished 16x128 staging tile to global with coalesced NT stores
    auto flush = [&](int tt) {
        float* obase = out + (size_t)(lstart + tt * TILE) * (D * E)
                           + (size_t)d0 * E;
        const float* src = ldsOut[tt & 1];
        #pragma unroll
        for (int j = 0; j < 2; ++j) {
            const int idx = tid + j * 256;   // v4f index in [0,512)
            const int row = idx >> 5;
            const int c4  = idx & 31;
            v4f v = *(const v4f*)&src[row * (16 * E) + c4 * 4];
            __builtin_nontemporal_store(
                v, (v4f*)(obase + (size_t)row * (D * E) + c4 * 4));
        }
    };

    // prefetch tile 0
    float xreg = x[(size_t)(lstart + xr) * D + d0 + xc];

    for (int t = 0; t < NTILES; ++t) {
        WG_BARRIER();                 // prior readers of ldsX / ldsOut done
        ldsX[tid] = xreg;             // stage tile t (waits its load here)
        if (t + 1 < NTILES)           // prefetch tile t+1; overlaps compute
            xreg = x[(size_t)(lstart + (t + 1) * TILE + xr) * D + d0 + xc];
        WG_BARRIER();                 // ldsX visible to all waves

        if (t > warm) flush(t - 1);   // NT-store previous tile during compute

        // accumulator pre-loaded with carry contribution a^{m+1} * c
        v8f acc;
        #pragma unroll
        for (int i = 0; i < 8; ++i) acc[i] = apc[i] * cval;

        // 16x16x16 GEMM as 4 chained f32 16x16x4 WMMAs
        #pragma unroll
        for (int kk = 0; kk < 4; ++kk) {
            v2f bM;
            bM.x = ldsX[(4 * kk + kb    ) * 16 + n];
            bM.y = ldsX[(4 * kk + kb + 1) * 16 + n];
            acc = __builtin_amdgcn_wmma_f32_16x16x4_f32(
                false, aM[kk], false, bM, (short)0, acc, false, false);
        }

        // next carry: row m=15 lives in acc[7] of lanes 16..31, column = lane-16
        cval = __shfl(acc[7], 16 + n, 32);

        if (t >= warm) {
            // scatter this wave's 16x16 tile into (l, d, e)-interleaved staging
            float* dst = ldsOut[t & 1];
            #pragma unroll
            for (int i = 0; i < 8; ++i)
                dst[(i + 8 * half) * (16 * E) + n * E + e] = acc[i];
        }
    }

    WG_BARRIER();
    flush(NTILES - 1);                // drain final staged tile
}

extern "C" void kernel_launch(void* const* d_in, const int* in_sizes, int n_in,
                              void* d_out, int out_size, void* d_ws, size_t ws_size,
                              hipStream_t stream) {
    (void)in_sizes; (void)n_in; (void)d_ws; (void)ws_size; (void)out_size;
    const float* x   = (const float*)d_in[0];
    const float* ldk = (const float*)d_in[1];
    float* out = (float*)d_out;

    dim3 grid(D / 16, NSEG);   // (32, 15)
    dim3 block(256);
    ema_scan_wmma<<<grid, block, 0, stream>>>(x, ldk, out);
}